// ExtrinsicsLearner_30374008717851
// MI455X (gfx1250) — compile-verified
//
#include <hip/hip_runtime.h>
#include <hip/hip_bf16.h>

typedef float v2f __attribute__((ext_vector_type(2)));
typedef float v4f __attribute__((ext_vector_type(4)));
typedef float v8f __attribute__((ext_vector_type(8)));

// Rigid inverse of c2w built from Rodrigues(r) and translation t.
// Matches reference's vec2skew convention (columns s0,s1,s2 => K = -K_standard).
__device__ __forceinline__ void compute_inv_c2w(float r0, float r1, float r2,
                                                float t0, float t1, float t2,
                                                float* Ei /*16 floats, row-major 4x4*/) {
  const float n2 = r0 * r0 + r1 * r1 + r2 * r2;
  const float n  = sqrtf(n2) + 1e-15f;
  const float a  = sinf(n) / n;
  const float b  = (1.0f - cosf(n)) / (n * n);
  // R = I + a*K + b*K^2 with K = [[0, r2, -r1], [-r2, 0, r0], [r1, -r0, 0]]
  // K^2 = r r^T - (r.r) I
  const float R00 = 1.0f - b * (r1 * r1 + r2 * r2);
  const float R01 =  a * r2 + b * r0 * r1;
  const float R02 = -a * r1 + b * r0 * r2;
  const float R10 = -a * r2 + b * r0 * r1;
  const float R11 = 1.0f - b * (r0 * r0 + r2 * r2);
  const float R12 =  a * r0 + b * r1 * r2;
  const float R20 =  a * r1 + b * r0 * r2;
  const float R21 = -a * r0 + b * r1 * r2;
  const float R22 = 1.0f - b * (r0 * r0 + r1 * r1);
  // inv = [[R^T, -R^T t], [0 0 0 1]]
  Ei[0]  = R00; Ei[1]  = R10; Ei[2]  = R20; Ei[3]  = -(R00 * t0 + R10 * t1 + R20 * t2);
  Ei[4]  = R01; Ei[5]  = R11; Ei[6]  = R21; Ei[7]  = -(R01 * t0 + R11 * t1 + R21 * t2);
  Ei[8]  = R02; Ei[9]  = R12; Ei[10] = R22; Ei[11] = -(R02 * t0 + R12 * t1 + R22 * t2);
  Ei[12] = 0.0f; Ei[13] = 0.0f; Ei[14] = 0.0f; Ei[15] = 1.0f;
}

// -------- Phase 1: per-frame table M[f] = E0[f] @ inv(c2w(f)) via WMMA --------
// Block = 128 threads = 4 waves; each wave handles 4 frames with one
// V_WMMA_F32_16X16X4_F32 (A = 4 stacked E0's, B = 4 side-by-side Einv's;
// diagonal 4x4 blocks of D are the products).
__global__ void __launch_bounds__(128)
build_table_wmma(const float* __restrict__ r, const float* __restrict__ t,
                 const float* __restrict__ E0, float* __restrict__ table, int nf) {
  __shared__ float sA[4][4][16];  // [wave][frame-in-wave][row*4+col] : E0
  __shared__ float sB[4][4][16];  // [wave][frame-in-wave][row*4+col] : inv(c2w)
  const int tid   = threadIdx.x;
  const int wave  = tid >> 5;
  const int lane  = tid & 31;
  const int fbase = blockIdx.x * 16 + wave * 4;

  if (lane < 4) {
    const int f = fbase + lane;
    float Ei[16], E[16];
    if (f < nf) {
      compute_inv_c2w(r[f * 3], r[f * 3 + 1], r[f * 3 + 2],
                      t[f * 3], t[f * 3 + 1], t[f * 3 + 2], Ei);
#pragma unroll
      for (int e = 0; e < 16; ++e) E[e] = E0[(size_t)f * 16 + e];
    } else {
#pragma unroll
      for (int e = 0; e < 16; ++e) { Ei[e] = (e % 5 == 0) ? 1.0f : 0.0f; E[e] = Ei[e]; }
    }
#pragma unroll
    for (int e = 0; e < 16; ++e) { sA[wave][lane][e] = E[e]; sB[wave][lane][e] = Ei[e]; }
  }
  __syncthreads();

  // f32 WMMA operand layout (ISA 7.12.2):
  //   A 16x4 : lanes 0-15 -> M=lane, VGPR0/1 = K0/K1 ; lanes 16-31 -> M=lane-16, K2/K3
  //   B 4x16 : lanes 0-15 -> N=lane, VGPR0/1 = K0/K1 ; lanes 16-31 -> N=lane-16, K2/K3
  //   D 16x16: lanes 0-15 -> VGPR g = (M=g, N=lane) ; lanes 16-31 -> (M=8+g, N=lane-16)
  const int half = lane >> 4;    // which K pair (A/B) / which M half (D)
  const int l    = lane & 15;    // M (A) / N (B,D)
  const int blk  = l >> 2;       // frame within this wave's group of 4
  const int sub  = l & 3;        // row within E0 block / col within Einv block
  const int k0   = half * 2;

  v2f A, B;
  A.x = sA[wave][blk][sub * 4 + k0];        // E0[blk][sub][k0]
  A.y = sA[wave][blk][sub * 4 + k0 + 1];    // E0[blk][sub][k0+1]
  B.x = sB[wave][blk][k0 * 4 + sub];        // Einv[blk][k0][sub]
  B.y = sB[wave][blk][(k0 + 1) * 4 + sub];  // Einv[blk][k0+1][sub]
  v8f C = {0.f, 0.f, 0.f, 0.f, 0.f, 0.f, 0.f, 0.f};
  v8f D = __builtin_amdgcn_wmma_f32_16x16x4_f32(false, A, false, B, (short)0, C,
                                                false, false);

  // Keep only the diagonal 4x4 blocks: D[4a:4a+4, 4a:4a+4] = E0_a @ Einv_a
#pragma unroll
  for (int g = 0; g < 8; ++g) {
    const int m = g + half * 8;
    if ((m >> 2) == blk) {
      const int f = fbase + blk;
      if (f < nf) table[(size_t)f * 16 + (m & 3) * 4 + sub] = D[g];
    }
  }
}

// -------- Phase 2: pure gather-copy (bandwidth bound, ~136 MB total) --------
// 4 threads per ray -> each wave stores 512B fully contiguous; NT stores keep
// the streaming output from evicting the L2-resident 640KB table.
__global__ void __launch_bounds__(256)
gather_rays(const int* __restrict__ cam, const v4f* __restrict__ table,
            v4f* __restrict__ out, int n4) {
  const int j = blockIdx.x * blockDim.x + threadIdx.x;
  if (j >= n4) return;
  const int ray = j >> 2;
  const int q   = j & 3;
  const int cid = cam[ray];
  v4f v = table[(size_t)cid * 4 + q];
  __builtin_nontemporal_store(v, out + j);
}

// -------- Fallback: fully fused per-ray path (if d_ws is too small) --------
__global__ void __launch_bounds__(256)
fused_rays(const float* __restrict__ r, const float* __restrict__ t,
           const float* __restrict__ E0, const int* __restrict__ cam,
           float* __restrict__ out, int nr) {
  const int i = blockIdx.x * blockDim.x + threadIdx.x;
  if (i >= nr) return;
  const int c = cam[i];
  float Ei[16];
  compute_inv_c2w(r[c * 3], r[c * 3 + 1], r[c * 3 + 2],
                  t[c * 3], t[c * 3 + 1], t[c * 3 + 2], Ei);
  const float* E = E0 + (size_t)c * 16;
  v4f* o = (v4f*)(out + (size_t)i * 16);
#pragma unroll
  for (int row = 0; row < 4; ++row) {
    const float e0 = E[row * 4 + 0], e1 = E[row * 4 + 1];
    const float e2 = E[row * 4 + 2], e3 = E[row * 4 + 3];
    v4f m;
#pragma unroll
    for (int col = 0; col < 4; ++col)
      m[col] = e0 * Ei[0 * 4 + col] + e1 * Ei[1 * 4 + col] +
               e2 * Ei[2 * 4 + col] + e3 * Ei[3 * 4 + col];
    __builtin_nontemporal_store(m, o + row);
  }
}

extern "C" void kernel_launch(void* const* d_in, const int* in_sizes, int n_in,
                              void* d_out, int out_size, void* d_ws, size_t ws_size,
                              hipStream_t stream) {
  const float* r   = (const float*)d_in[0];  // (NF, 3)
  const float* t   = (const float*)d_in[1];  // (NF, 3)
  const float* E0  = (const float*)d_in[2];  // (NF, 4, 4)
  const int*   cam = (const int*)d_in[3];    // (NR,)
  const int nf = in_sizes[0] / 3;
  const int nr = in_sizes[3];
  float* out = (float*)d_out;

  const size_t need = (size_t)nf * 16 * sizeof(float);
  if (d_ws && ws_size >= need) {
    float* table = (float*)d_ws;
    build_table_wmma<<<(nf + 15) / 16, 128, 0, stream>>>(r, t, E0, table, nf);
    const int n4 = nr * 4;
    gather_rays<<<(n4 + 255) / 256, 256, 0, stream>>>(cam, (const v4f*)table,
                                                      (v4f*)out, n4);
  } else {
    fused_rays<<<(nr + 255) / 256, 256, 0, stream>>>(r, t, E0, cam, out, nr);
  }
}